// Net_37598143709632
// MI455X (gfx1250) — compile-verified
//
#include <hip/hip_runtime.h>
#include <math.h>

// ---- problem constants (from reference) ----
#define IN_DIM   128
#define HEADS1   4
#define HID      64
#define C1       (HEADS1 * HID)   // 256
#define OUT_DIM  12
#define NUM_GRAPHS 512
#define NEG_SLOPE 0.2f

typedef __attribute__((ext_vector_type(2))) float v2f;
typedef __attribute__((ext_vector_type(8))) float v8f;

// ------------------------------------------------------------------
// utility: fill buffer with constant (used to init accumulators each call)
// ------------------------------------------------------------------
__global__ void gat_fill(float* __restrict__ p, float v, long long n) {
  long long i = (long long)blockIdx.x * blockDim.x + threadIdx.x;
  long long stride = (long long)gridDim.x * blockDim.x;
  for (; i < n; i += stride) p[i] = v;
}

// ------------------------------------------------------------------
// fp32 WMMA GEMM: C[M,N] = A[M,K] * B[K,N]   (row-major, fp32 end to end)
// One wave -> 16x64 output strip via 4 accumulators; 8 waves per block.
// Uses V_WMMA_F32_16X16X4_F32 (the CDNA5 fp32 tensor path).
// A frag (16x4):  lane holds M = lane&15, K = k0 + 2*(lane>>4) + r  in v[r]
// B frag (4x16):  lane holds N = lane&15, K = k0 + 2*(lane>>4) + r  in v[r]
// C/D (16x16):    VGPR i: M = i + 8*(lane>>4), N = lane&15
// ------------------------------------------------------------------
__global__ __launch_bounds__(256) void gat_wmma_gemm_f32(
    const float* __restrict__ A, const float* __restrict__ B,
    float* __restrict__ C, int M, int N, int K)
{
  const int lane  = threadIdx.x & 31;
  const int wave  = threadIdx.x >> 5;
  const int mtile = blockIdx.x * 8 + wave;
  const int m0    = mtile * 16;
  if (m0 >= M) return;
  const int n0    = blockIdx.y * 64;
  const int mrow  = lane & 15;   // M for A, N for B/C
  const int khalf = lane >> 4;   // 0 or 1

  v8f acc0 = {}, acc1 = {}, acc2 = {}, acc3 = {};
  const float* Arow = A + (size_t)(m0 + mrow) * K;

  for (int k0 = 0; k0 < K; k0 += 4) {
    const int ka = k0 + 2 * khalf;
    v2f a, b0, b1, b2, b3;
    a[0] = Arow[ka];
    a[1] = Arow[ka + 1];
    const float* Br0 = B + (size_t)ka * N + n0 + mrow;       // row ka
    const float* Br1 = Br0 + N;                              // row ka+1
    b0[0] = Br0[0];  b0[1] = Br1[0];
    b1[0] = Br0[16]; b1[1] = Br1[16];
    b2[0] = Br0[32]; b2[1] = Br1[32];
    b3[0] = Br0[48]; b3[1] = Br1[48];
    acc0 = __builtin_amdgcn_wmma_f32_16x16x4_f32(false, a, false, b0, (short)0, acc0, false, false);
    acc1 = __builtin_amdgcn_wmma_f32_16x16x4_f32(false, a, false, b1, (short)0, acc1, false, false);
    acc2 = __builtin_amdgcn_wmma_f32_16x16x4_f32(false, a, false, b2, (short)0, acc2, false, false);
    acc3 = __builtin_amdgcn_wmma_f32_16x16x4_f32(false, a, false, b3, (short)0, acc3, false, false);
  }

  const int ccol = n0 + mrow;
  for (int i = 0; i < 8; ++i) {
    float* Crow = C + (size_t)(m0 + i + 8 * khalf) * N + ccol;
    Crow[0]  = acc0[i];
    Crow[16] = acc1[i];
    Crow[32] = acc2[i];
    Crow[48] = acc3[i];
  }
}

// ------------------------------------------------------------------
// per-(node,head) attention dot products: as[n,h] = <h[n,h,:], a_src[h,:]>
// dim is a multiple of 4 -> float4 reads
// ------------------------------------------------------------------
__global__ void gat_alpha_dots(const float* __restrict__ h,
                               const float* __restrict__ a_src,
                               const float* __restrict__ a_dst,
                               float* __restrict__ as_, float* __restrict__ ad_,
                               int n, int heads, int dim)
{
  int i = blockIdx.x * blockDim.x + threadIdx.x;
  if (i >= n * heads) return;
  int node = i / heads, hh = i % heads;
  const float4* hp = (const float4*)(h + (size_t)node * heads * dim + hh * dim);
  const float4* sp = (const float4*)(a_src + hh * dim);
  const float4* dp = (const float4*)(a_dst + hh * dim);
  float s = 0.f, d = 0.f;
  for (int k = 0; k < dim / 4; ++k) {
    float4 v = hp[k], a = sp[k], b = dp[k];
    s += v.x * a.x + v.y * a.y + v.z * a.z + v.w * a.w;
    d += v.x * b.x + v.y * b.y + v.z * b.z + v.w * b.w;
  }
  as_[i] = s;
  ad_[i] = d;
}

// float atomic max via monotonic int transform (valid: init = -inf, no NaNs)
__device__ __forceinline__ void atomicMaxF(float* addr, float val) {
  if (val >= 0.0f) atomicMax((int*)addr, __float_as_int(val));
  else             atomicMin((unsigned int*)addr, __float_as_uint(val));
}

// edge pass 1: segment max of leaky_relu logits into m[dst,h]
__global__ void gat_edge_max(const int* __restrict__ src, const int* __restrict__ dst,
                             int E, int n_nodes,
                             const float* __restrict__ as_, const float* __restrict__ ad_,
                             float* __restrict__ m, int heads)
{
  int e = blockIdx.x * blockDim.x + threadIdx.x;
  if (e >= E + n_nodes) return;
  int s = (e < E) ? src[e] : (e - E);   // self-loops appended
  int d = (e < E) ? dst[e] : (e - E);
  for (int h = 0; h < heads; ++h) {
    float v = as_[s * heads + h] + ad_[d * heads + h];
    v = v > 0.f ? v : NEG_SLOPE * v;
    atomicMaxF(&m[d * heads + h], v);
  }
}

// edge pass 2: w = exp(logit - m[dst]); store w, accumulate z[dst]
__global__ void gat_edge_weight(const int* __restrict__ src, const int* __restrict__ dst,
                                int E, int n_nodes,
                                const float* __restrict__ as_, const float* __restrict__ ad_,
                                const float* __restrict__ m,
                                float* __restrict__ z, float* __restrict__ we, int heads)
{
  int e = blockIdx.x * blockDim.x + threadIdx.x;
  if (e >= E + n_nodes) return;
  int s = (e < E) ? src[e] : (e - E);
  int d = (e < E) ? dst[e] : (e - E);
  for (int h = 0; h < heads; ++h) {
    float v = as_[s * heads + h] + ad_[d * heads + h];
    v = v > 0.f ? v : NEG_SLOPE * v;
    float w = __expf(v - m[d * heads + h]);
    we[(size_t)e * heads + h] = w;
    atomicAdd(&z[d * heads + h], w);
  }
}

// edge pass 3: agg[dst,c..c+3] += w[e,h(c)] * h[src,c..c+3]
// one thread per (edge, 4-channel group): b128 gather of h, 4 f32 atomics
// (4 consecutive channels always share one head since dim % 4 == 0)
__global__ void gat_edge_scatter4(const int* __restrict__ src, const int* __restrict__ dst,
                                  int E, int n_nodes,
                                  const float* __restrict__ we, const float* __restrict__ h,
                                  float* __restrict__ agg, int heads, int dim)
{
  const int C  = heads * dim;
  const int CV = C / 4;
  long long gid = (long long)blockIdx.x * blockDim.x + threadIdx.x;
  long long total = (long long)(E + n_nodes) * CV;
  if (gid >= total) return;
  int cv = (int)(gid % CV);
  int e  = (int)(gid / CV);
  int c  = cv * 4;
  int s = (e < E) ? src[e] : (e - E);
  int d = (e < E) ? dst[e] : (e - E);
  float w = we[(size_t)e * heads + c / dim];
  float4 hv = *(const float4*)(h + (size_t)s * C + c);
  float* ap = agg + (size_t)d * C + c;
  atomicAdd(ap + 0, w * hv.x);
  atomicAdd(ap + 1, w * hv.y);
  atomicAdd(ap + 2, w * hv.z);
  atomicAdd(ap + 3, w * hv.w);
}

// node epilogue: out = elu(agg / (z + eps) + bias)   (in place, b128 I/O)
__global__ void gat_node_finalize4(float* __restrict__ agg, const float* __restrict__ z,
                                   const float* __restrict__ b, int n, int heads, int dim)
{
  const int C  = heads * dim;
  const int CV = C / 4;
  long long gid = (long long)blockIdx.x * blockDim.x + threadIdx.x;
  if (gid >= (long long)n * CV) return;
  int cv   = (int)(gid % CV);
  int node = (int)(gid / CV);
  int c    = cv * 4;
  float zz = z[node * heads + c / dim] + 1e-16f;
  float4 v  = *(float4*)(agg + (size_t)node * C + c);
  float4 bb = *(const float4*)(b + c);
  float o0 = v.x / zz + bb.x;
  float o1 = v.y / zz + bb.y;
  float o2 = v.z / zz + bb.z;
  float o3 = v.w / zz + bb.w;
  float4 r;
  r.x = o0 > 0.f ? o0 : expm1f(o0);
  r.y = o1 > 0.f ? o1 : expm1f(o1);
  r.z = o2 > 0.f ? o2 : expm1f(o2);
  r.w = o3 > 0.f ? o3 : expm1f(o3);
  *(float4*)(agg + (size_t)node * C + c) = r;
}

// global add pool (b128 gather, 4 f32 atomics)
__global__ void gat_pool4(const float* __restrict__ h, const int* __restrict__ batch,
                          float* __restrict__ pooled, int n, int dim)
{
  const int DV = dim / 4;
  long long gid = (long long)blockIdx.x * blockDim.x + threadIdx.x;
  if (gid >= (long long)n * DV) return;
  int cv   = (int)(gid % DV);
  int node = (int)(gid / DV);
  int c    = cv * 4;
  float4 hv = *(const float4*)(h + (size_t)node * dim + c);
  float* pp = pooled + (size_t)batch[node] * dim + c;
  atomicAdd(pp + 0, hv.x);
  atomicAdd(pp + 1, hv.y);
  atomicAdd(pp + 2, hv.z);
  atomicAdd(pp + 3, hv.w);
}

// final linear: out[512,12] = pooled[512,64] @ Wl[64,12] + bl
__global__ void gat_final_linear(const float* __restrict__ pooled,
                                 const float* __restrict__ Wl,
                                 const float* __restrict__ bl,
                                 float* __restrict__ out)
{
  int i = blockIdx.x * blockDim.x + threadIdx.x;
  if (i >= NUM_GRAPHS * OUT_DIM) return;
  int g = i / OUT_DIM, o = i % OUT_DIM;
  float acc = bl[o];
  for (int d = 0; d < HID; ++d) acc += pooled[g * HID + d] * Wl[d * OUT_DIM + o];
  out[i] = acc;
}

// ------------------------------------------------------------------
extern "C" void kernel_launch(void* const* d_in, const int* in_sizes, int n_in,
                              void* d_out, int out_size, void* d_ws, size_t ws_size,
                              hipStream_t stream)
{
  const float* x      = (const float*)d_in[0];
  const int*   ei     = (const int*)  d_in[1];
  const int*   batch  = (const int*)  d_in[2];
  const float* W1     = (const float*)d_in[3];
  const float* a_src1 = (const float*)d_in[4];
  const float* a_dst1 = (const float*)d_in[5];
  const float* b1     = (const float*)d_in[6];
  const float* W2     = (const float*)d_in[7];
  const float* a_src2 = (const float*)d_in[8];
  const float* a_dst2 = (const float*)d_in[9];
  const float* b2     = (const float*)d_in[10];
  const float* Wl     = (const float*)d_in[11];
  const float* bl     = (const float*)d_in[12];
  float* out = (float*)d_out;

  const int Nn = in_sizes[0] / IN_DIM;   // 50000
  const int E  = in_sizes[1] / 2;        // 400000
  const int ET = E + Nn;                 // edges incl. self-loops
  const int* src = ei;                   // edge_index row 0
  const int* dst = ei + E;               // edge_index row 1

  // workspace layout (floats)
  float* ws = (float*)d_ws;
  size_t off = 0;
  float* h1   = ws + off; off += (size_t)Nn * C1;       // conv1 features
  float* as1  = ws + off; off += (size_t)Nn * HEADS1;
  float* ad1  = ws + off; off += (size_t)Nn * HEADS1;
  float* m1   = ws + off; off += (size_t)Nn * HEADS1;
  float* z1   = ws + off; off += (size_t)Nn * HEADS1;
  float* w1e  = ws + off; off += (size_t)ET * HEADS1;
  float* agg1 = ws + off; off += (size_t)Nn * C1;
  float* h2   = ws + off; off += (size_t)Nn * HID;
  float* as2  = ws + off; off += (size_t)Nn;
  float* ad2  = ws + off; off += (size_t)Nn;
  float* m2   = ws + off; off += (size_t)Nn;
  float* z2   = ws + off; off += (size_t)Nn;
  float* w2e  = ws + off; off += (size_t)ET;
  float* agg2 = ws + off; off += (size_t)Nn * HID;
  float* pooled = ws + off; off += (size_t)NUM_GRAPHS * HID;
  (void)ws_size; (void)n_in; (void)out_size;

  const int T = 256;
  const float NEG_INF = -INFINITY;
  // init accumulators (fresh every call; harness does not re-poison)
  gat_fill<<<1024, T, 0, stream>>>(m1, NEG_INF, (long long)Nn * HEADS1);
  gat_fill<<<1024, T, 0, stream>>>(z1, 0.f,     (long long)Nn * HEADS1);
  gat_fill<<<4096, T, 0, stream>>>(agg1, 0.f,   (long long)Nn * C1);
  gat_fill<<<1024, T, 0, stream>>>(m2, NEG_INF, (long long)Nn);
  gat_fill<<<1024, T, 0, stream>>>(z2, 0.f,     (long long)Nn);
  gat_fill<<<2048, T, 0, stream>>>(agg2, 0.f,   (long long)Nn * HID);
  gat_fill<<<128,  T, 0, stream>>>(pooled, 0.f, (long long)NUM_GRAPHS * HID);

  const int mtiles = (Nn + 15) / 16;            // 3125
  const int mblocks = (mtiles + 7) / 8;         // 8 waves (row tiles) per block

  // ---- layer 1 ----
  gat_wmma_gemm_f32<<<dim3(mblocks, C1 / 64), T, 0, stream>>>(x, W1, h1, Nn, C1, IN_DIM);
  gat_alpha_dots<<<(Nn * HEADS1 + T - 1) / T, T, 0, stream>>>(h1, a_src1, a_dst1, as1, ad1, Nn, HEADS1, HID);
  gat_edge_max   <<<(ET + T - 1) / T, T, 0, stream>>>(src, dst, E, Nn, as1, ad1, m1, HEADS1);
  gat_edge_weight<<<(ET + T - 1) / T, T, 0, stream>>>(src, dst, E, Nn, as1, ad1, m1, z1, w1e, HEADS1);
  {
    long long tot = (long long)ET * (C1 / 4);
    gat_edge_scatter4<<<(unsigned)((tot + T - 1) / T), T, 0, stream>>>(src, dst, E, Nn, w1e, h1, agg1, HEADS1, HID);
  }
  gat_node_finalize4<<<(unsigned)(((long long)Nn * (C1 / 4) + T - 1) / T), T, 0, stream>>>(agg1, z1, b1, Nn, HEADS1, HID);

  // ---- layer 2 (heads=1, concat=False -> mean over 1 head == identity) ----
  gat_wmma_gemm_f32<<<dim3(mblocks, HID / 64), T, 0, stream>>>(agg1, W2, h2, Nn, HID, C1);
  gat_alpha_dots<<<(Nn + T - 1) / T, T, 0, stream>>>(h2, a_src2, a_dst2, as2, ad2, Nn, 1, HID);
  gat_edge_max   <<<(ET + T - 1) / T, T, 0, stream>>>(src, dst, E, Nn, as2, ad2, m2, 1);
  gat_edge_weight<<<(ET + T - 1) / T, T, 0, stream>>>(src, dst, E, Nn, as2, ad2, m2, z2, w2e, 1);
  {
    long long tot = (long long)ET * (HID / 4);
    gat_edge_scatter4<<<(unsigned)((tot + T - 1) / T), T, 0, stream>>>(src, dst, E, Nn, w2e, h2, agg2, 1, HID);
  }
  gat_node_finalize4<<<(unsigned)(((long long)Nn * (HID / 4) + T - 1) / T), T, 0, stream>>>(agg2, z2, b2, Nn, 1, HID);

  // ---- pool + head ----
  gat_pool4<<<(unsigned)(((long long)Nn * (HID / 4) + T - 1) / T), T, 0, stream>>>(agg2, batch, pooled, Nn, HID);
  gat_final_linear<<<(NUM_GRAPHS * OUT_DIM + T - 1) / T, T, 0, stream>>>(pooled, Wl, bl, out);
}